// Abbott_54889682042972
// MI455X (gfx1250) — compile-verified
//
#include <hip/hip_runtime.h>

typedef __attribute__((ext_vector_type(2))) float v2f;
typedef __attribute__((ext_vector_type(4))) float f4;
typedef __attribute__((ext_vector_type(8))) float v8f;

// Problem constants (match reference)
constexpr int   B_      = 8;
constexpr int   N_      = 2048;
constexpr float ALPHA_P = 0.95f;
constexpr float ALPHA_D = 0.9f;
constexpr float WMAX    = 1.0f;

// Tiling
constexpr int THREADS   = 256;
constexpr int VEC       = 4;
constexpr int O_PER_BLK = THREADS * VEC;   // 1024 o-values per block
constexpr int OT        = N_ / O_PER_BLK;  // 2 o-tiles
constexpr int CHUNKS    = 32;              // e-dimension split for parallelism
constexpr int E_PER     = N_ / CHUNKS;     // 64 e-rows per chunk

__device__ __forceinline__ f4 clampf4(f4 x, float lo, float hi) {
    f4 r;
    r.x = fminf(fmaxf(x.x, lo), hi);
    r.y = fminf(fmaxf(x.y, lo), hi);
    r.z = fminf(fmaxf(x.z, lo), hi);
    r.w = fminf(fmaxf(x.w, lo), hi);
    return r;
}

__device__ __forceinline__ f4 ldg_nt(const float* p) {
    return __builtin_nontemporal_load((const f4*)p);
}
__device__ __forceinline__ void stg_nt(float* p, f4 v) {
    __builtin_nontemporal_store(v, (f4*)p);
}

// ---------------------------------------------------------------------------
// Kernel 1: single streaming pass over the big [B,N,N] tensors.
// Produces W_new, xbar_pre_new, xbar_post_new, and per-chunk partial sums of
// I[b,o] = sum_e Xpre[b,e,o]*W[b,e,o] into workspace (deterministic order).
// ---------------------------------------------------------------------------
__global__ __launch_bounds__(THREADS)
void abbott_stream_kernel(const float* __restrict__ Xpre,
                          const float* __restrict__ Xpost,
                          const float* __restrict__ W,
                          const float* __restrict__ xbar_pre,
                          const float* __restrict__ xbar_post,
                          const float* __restrict__ Ap,
                          const float* __restrict__ Ad,
                          float* __restrict__ out_xbar_pre,
                          float* __restrict__ out_xbar_post,
                          float* __restrict__ out_W,
                          float* __restrict__ ws_partial)
{
    const int bid   = blockIdx.x;
    const int otile = bid % OT;
    const int b     = (bid / OT) % B_;
    const int chunk = bid / (OT * B_);
    const int o     = otile * O_PER_BLK + (int)threadIdx.x * VEC;

    // Per-output-neuron scalars for this batch (broadcast over e)
    const size_t bo = (size_t)b * N_ + o;
    const f4 xp  = *(const f4*)(Xpost     + bo);
    const f4 xbp = *(const f4*)(xbar_post + bo);

    f4 acc = (f4)0.0f;

    const int e0 = chunk * E_PER;
    size_t base  = ((size_t)b * N_ + e0) * N_ + o;   // index into [B,N,N]
    size_t abase = (size_t)e0 * N_ + o;              // index into [N,N]

    #pragma unroll 4
    for (int i = 0; i < E_PER; ++i) {
        // Stream-once tensors: non-temporal (don't pollute L2; A_p/A_d stay hot)
        const f4 xpre  = ldg_nt(Xpre     + base);
        const f4 w     = ldg_nt(W        + base);
        const f4 xbpre = ldg_nt(xbar_pre + base);
        const f4 ap    = *(const f4*)(Ap + abase);   // reused 8x per b -> L2 hit
        const f4 ad    = *(const f4*)(Ad + abase);

        acc += xpre * w;                                  // I partial

        const f4 dw   = xp * xbpre * ap - xpre * xbp * ad;
        const f4 wn   = clampf4(w + dw, 0.0f, WMAX);
        const f4 xbn  = ALPHA_P * xbpre + (1.0f - ALPHA_P) * xpre;

        stg_nt(out_W        + base, wn);
        stg_nt(out_xbar_pre + base, xbn);

        base  += N_;
        abase += N_;
    }

    // One partial vector per (chunk, b, o) -> workspace [CHUNKS][B*N]
    *(f4*)(ws_partial + (size_t)chunk * (B_ * N_) + bo) = acc;

    if (chunk == 0) {
        *(f4*)(out_xbar_post + bo) = ALPHA_D * xbp + (1.0f - ALPHA_D) * xp;
    }
}

// ---------------------------------------------------------------------------
// Kernel 2: reduce the CHUNKS partial sums per output with V_WMMA_F32_16X16X4_F32.
// A = ones(16x4); D[m][n] = sum_k B[k][n] + C[m][n]  -> every D row holds the
// column sums, independent of the K<->slot mapping (A is all-ones), so this is
// layout-robust. One wave (32 lanes, EXEC all-1s) reduces 16 outputs with 8
// chained k=4 WMMAs. Fully deterministic (no float atomics).
// ---------------------------------------------------------------------------
__global__ __launch_bounds__(32)
void abbott_reduce_wmma(const float* __restrict__ ws_partial,
                        float* __restrict__ I_out)
{
    const int lane   = threadIdx.x;   // 0..31
    const int half   = lane >> 4;     // 0 or 1  (selects K row within a VGPR)
    const int n      = lane & 15;     // output column within the 16-wide tile
    const int o_base = blockIdx.x * 16;  // flat index into [B*N]

    v2f a; a.x = 1.0f; a.y = 1.0f;    // A = ones(16x4) in every lane/slot
    v8f c = {};

    #pragma unroll
    for (int s = 0; s < CHUNKS / 4; ++s) {
        // Four distinct chunks occupy the four (vgpr, lane-half) K-slots.
        v2f bm;
        bm.x = ws_partial[(size_t)(4 * s + half)     * (B_ * N_) + o_base + n];
        bm.y = ws_partial[(size_t)(4 * s + 2 + half) * (B_ * N_) + o_base + n];
        // 8 args: (neg_a, A, neg_b, B, c_mod, C, reuse_a, reuse_b)
        c = __builtin_amdgcn_wmma_f32_16x16x4_f32(
                /*neg_a=*/false, a, /*neg_b=*/false, bm,
                /*c_mod=*/(short)0, c, /*reuse_a=*/false, /*reuse_b=*/false);
    }

    // Row M=0 of D lives in c[0], lanes 0..15 (N = lane). All rows are equal.
    if (lane < 16) {
        I_out[o_base + lane] = c[0];
    }
}

// ---------------------------------------------------------------------------
extern "C" void kernel_launch(void* const* d_in, const int* in_sizes, int n_in,
                              void* d_out, int out_size, void* d_ws, size_t ws_size,
                              hipStream_t stream)
{
    (void)in_sizes; (void)n_in; (void)out_size; (void)ws_size;

    const float* Xpre      = (const float*)d_in[0];
    const float* Xpost     = (const float*)d_in[1];
    const float* W         = (const float*)d_in[2];
    const float* xbar_pre  = (const float*)d_in[3];
    const float* xbar_post = (const float*)d_in[4];
    const float* Ap        = (const float*)d_in[5];
    const float* Ad        = (const float*)d_in[6];

    // Outputs concatenated flat in reference return order.
    float* out            = (float*)d_out;
    float* I_out          = out;                                   // [B,N]
    float* out_xbar_pre   = I_out + (size_t)B_ * N_;               // [B,N,N]
    float* out_xbar_post  = out_xbar_pre + (size_t)B_ * N_ * N_;   // [B,N]
    float* out_W          = out_xbar_post + (size_t)B_ * N_;       // [B,N,N]

    float* ws = (float*)d_ws;  // [CHUNKS][B*N] partials = 2 MB

    const dim3 g1(OT * B_ * CHUNKS);  // 512 blocks
    abbott_stream_kernel<<<g1, THREADS, 0, stream>>>(
        Xpre, Xpost, W, xbar_pre, xbar_post, Ap, Ad,
        out_xbar_pre, out_xbar_post, out_W, ws);

    const dim3 g2((B_ * N_) / 16);    // 1024 waves, 16 outputs each
    abbott_reduce_wmma<<<g2, 32, 0, stream>>>(ws, I_out);
}